// MoEHead_28192165330993
// MI455X (gfx1250) — compile-verified
//
#include <hip/hip_runtime.h>

// MoE head: out[b,c] = sum_k softmax(emb@Wg+bg)[b,k] * relu(emb@We[k]+be[k])[c]
// B=131072 D=512 C=5 K=10. Memory-bound: 256MB emb stream @23.3TB/s ~= 11us.
// Strategy: one fused [B,512]x[512,64] fp32 WMMA GEMM (cols 0..9 = gate logits,
// 10..59 = expert logits, 60..63 = zero pad), weights staged in LDS, epilogue
// fused per 16-row tile.

#define D_DIM   512
#define C_DIM   5
#define K_EXP   10
#define NPAD    64
#define NCOL    60          // 10 gate + 50 expert columns
#define WPITCH  516         // LDS pitch for Wt[n][k]: bank = 4n -> conflict-free
#define SPITCH  68          // LDS pitch for epilogue scratch
#define BLOCKS  512
#define THREADS 256
#define WAVES   8           // wave32: 256 threads = 8 waves
#define TILES_PER_WAVE 2    // 8192 M-tiles / (512*8) waves

typedef __attribute__((ext_vector_type(2))) float v2f;
typedef __attribute__((ext_vector_type(4))) float v4f;
typedef __attribute__((ext_vector_type(8))) float v8f;

extern __shared__ float smem[];

__global__ __launch_bounds__(THREADS)
void moe_head_wmma_kernel(const float* __restrict__ emb,
                          const float* __restrict__ We,
                          const float* __restrict__ be,
                          const float* __restrict__ Wg,
                          const float* __restrict__ bg,
                          float* __restrict__ out) {
    float* ldsW = smem;                          // [64][WPITCH]  combined W^T
    float* scr  = smem + NPAD * WPITCH;          // [WAVES][16][SPITCH]
    float* bias = scr + WAVES * 16 * SPITCH;     // [64]

    const int tid  = threadIdx.x;
    const int lane = tid & 31;
    const int wave = tid >> 5;
    const int half = lane >> 4;    // 0: K offset 0,1   1: K offset 2,3
    const int l15  = lane & 15;

    // ---- stage combined weight matrix transposed into LDS: Wt[n][k] ----
    for (int e = tid; e < NPAD * D_DIM; e += THREADS) {
        int n = e & (NPAD - 1);
        int k = e >> 6;
        float v = 0.0f;
        if (n < K_EXP) {
            v = Wg[k * K_EXP + n];                                   // gate col
        } else if (n < NCOL) {
            int q = n - K_EXP;                                       // = k*5+c
            v = We[(q / C_DIM) * (D_DIM * C_DIM) + k * C_DIM + (q % C_DIM)];
        }
        ldsW[n * WPITCH + k] = v;
    }
    if (tid < NPAD) {
        float v = 0.0f;
        if (tid < K_EXP)      v = bg[tid];
        else if (tid < NCOL)  v = be[tid - K_EXP];   // be flat index = k*C+c
        bias[tid] = v;
    }
    __syncthreads();

    const int gwave  = blockIdx.x * WAVES + wave;
    float*    myscr  = scr + wave * 16 * SPITCH;

    for (int it = 0; it < TILES_PER_WAVE; ++it) {
        const int tile = gwave * TILES_PER_WAVE + it;   // 16-row M tile
        // A fragment base: row = tile*16 + (lane&15), K sub-offset = 2*(lane>>4)
        const float* arow = emb + (size_t)(tile * 16 + l15) * D_DIM + half * 2;
        // B fragment base in LDS (per N-tile add t*16*WPITCH)
        const float* wrow = ldsW + l15 * WPITCH + half * 2;

        v8f acc0 = {}, acc1 = {}, acc2 = {}, acc3 = {};

        for (int k0 = 0; k0 < D_DIM; k0 += 16) {
            // 4 k-steps of A per chunk: one float2 per step per lane
            v2f a0 = *(const v2f*)(arow + k0 + 0);
            v2f a1 = *(const v2f*)(arow + k0 + 4);
            v2f a2 = *(const v2f*)(arow + k0 + 8);
            v2f a3 = *(const v2f*)(arow + k0 + 12);
#pragma unroll
            for (int j = 0; j < 4; ++j) {
                v2f a = (j == 0) ? a0 : (j == 1) ? a1 : (j == 2) ? a2 : a3;
                const float* wk = wrow + k0 + j * 4;
                v2f b;
                b = *(const v2f*)(wk + 0 * 16 * WPITCH);
                acc0 = __builtin_amdgcn_wmma_f32_16x16x4_f32(
                           false, a, false, b, (short)0, acc0, false, false);
                b = *(const v2f*)(wk + 1 * 16 * WPITCH);
                acc1 = __builtin_amdgcn_wmma_f32_16x16x4_f32(
                           false, a, false, b, (short)0, acc1, false, false);
                b = *(const v2f*)(wk + 2 * 16 * WPITCH);
                acc2 = __builtin_amdgcn_wmma_f32_16x16x4_f32(
                           false, a, false, b, (short)0, acc2, false, false);
                b = *(const v2f*)(wk + 3 * 16 * WPITCH);
                acc3 = __builtin_amdgcn_wmma_f32_16x16x4_f32(
                           false, a, false, b, (short)0, acc3, false, false);
            }
        }

        // ---- spill 16x64 logit tile to LDS scratch ----
        // C/D layout: VGPR j holds row (j + 8*half), col = l15 (+16*t)
#pragma unroll
        for (int j = 0; j < 8; ++j) {
            int r = j + 8 * half;
            myscr[r * SPITCH +  0 + l15] = acc0[j];
            myscr[r * SPITCH + 16 + l15] = acc1[j];
            myscr[r * SPITCH + 32 + l15] = acc2[j];
            myscr[r * SPITCH + 48 + l15] = acc3[j];
        }
        asm volatile("s_wait_dscnt 0x0" ::: "memory");

        // ---- fused epilogue: lanes 0..15 each own one row ----
        if (lane < 16) {
            float l[64];
#pragma unroll
            for (int i = 0; i < 16; ++i) {
                v4f v = *(const v4f*)(myscr + lane * SPITCH + i * 4);
#pragma unroll
                for (int q = 0; q < 4; ++q) l[i * 4 + q] = v[q];
            }
#pragma unroll
            for (int i = 0; i < NCOL; ++i) l[i] += bias[i];

            // softmax over gate logits l[0..9]
            float gmax = l[0];
#pragma unroll
            for (int k = 1; k < K_EXP; ++k) gmax = fmaxf(gmax, l[k]);
            float ge[K_EXP];
            float gsum = 0.0f;
#pragma unroll
            for (int k = 0; k < K_EXP; ++k) {
                ge[k] = __expf(l[k] - gmax);
                gsum += ge[k];
            }
            // gate-weighted relu(expert logits)
            float o[C_DIM] = {0.f, 0.f, 0.f, 0.f, 0.f};
#pragma unroll
            for (int k = 0; k < K_EXP; ++k) {
#pragma unroll
                for (int c = 0; c < C_DIM; ++c) {
                    o[c] += ge[k] * fmaxf(l[K_EXP + k * C_DIM + c], 0.0f);
                }
            }
            float inv = 1.0f / gsum;
            float* orow = out + (size_t)(tile * 16 + lane) * C_DIM;
#pragma unroll
            for (int c = 0; c < C_DIM; ++c) orow[c] = o[c] * inv;
        }
    }
}

extern "C" void kernel_launch(void* const* d_in, const int* in_sizes, int n_in,
                              void* d_out, int out_size, void* d_ws, size_t ws_size,
                              hipStream_t stream) {
    const float* emb = (const float*)d_in[0];   // [B, D]
    const float* We  = (const float*)d_in[1];   // [K, D, C]
    const float* be  = (const float*)d_in[2];   // [K, C]
    const float* Wg  = (const float*)d_in[3];   // [D, K]
    const float* bg  = (const float*)d_in[4];   // [K]
    float*       out = (float*)d_out;           // [B, C]

    const size_t lds_bytes =
        (size_t)(NPAD * WPITCH + WAVES * 16 * SPITCH + NPAD) * sizeof(float);

    moe_head_wmma_kernel<<<dim3(BLOCKS), dim3(THREADS), lds_bytes, stream>>>(
        emb, We, be, Wg, bg, out);
}